// SAGEMLP_70033736728588
// MI455X (gfx1250) — compile-verified
//
#include <hip/hip_runtime.h>
#include <hip/hip_bf16.h>

// ---------------------------------------------------------------------------
// SAGE(gcn) x2 + edge MLP predictor, restructured so all dense math runs on
// V_WMMA_F32_16X16X4_F32 and per-edge work is O(16) instead of O(256*16).
//   y   = x @ W1                       (WMMA, K=128)
//   h1  = relu((scatter(y)+y)/(deg+1) + b1)
//   agg = (scatter(h1)+h1)/(deg+1)
//   h2  = agg @ W2 + b2                (WMMA, K=16)
//   su  = h2 @ Wmlp[0:128], sv = h2 @ Wmlp[128:256]   (WMMA, K=128, shared A)
//   out[e] = su[src[e]] + sv[dst[e]] + bmlp
// ---------------------------------------------------------------------------

typedef __attribute__((ext_vector_type(2))) float v2f;
typedef __attribute__((ext_vector_type(8))) float v8f;

// fp32 WMMA 16x16x4: A 16x4 (lane<16: M=lane,K=0..1; lane>=16: M=lane-16,K=2..3),
// B 4x16 mirrored (lane owns column N=lane&15, K pair selected by lane>>4),
// C/D: VGPR v -> row M = v + 8*(lane>>4), col N = lane&15.
__device__ __forceinline__ v8f wmma_f32(v2f a, v2f b, v8f c) {
    return __builtin_amdgcn_wmma_f32_16x16x4_f32(false, a, false, b, (short)0, c,
                                                 false, false);
}

// ---------------- GEMM 1: y[N,16] = x[N,128] @ W1[128,16] -------------------
__global__ __launch_bounds__(256)
void k_gemm_xW1(const float* __restrict__ x, const float* __restrict__ W1,
                float* __restrict__ y, int nRows) {
    int wave = (blockIdx.x * blockDim.x + threadIdx.x) >> 5;   // 1 row-tile / wave
    int lane = threadIdx.x & 31;
    int nTiles = (nRows + 15) >> 4;
    if (wave >= nTiles) return;                                 // wave-uniform exit
    int row0  = wave << 4;
    bool full = (row0 + 16 <= nRows);                           // wave-uniform
    int mr    = lane & 15;
    int khalf = (lane >> 4) << 1;                               // 0 or 2
    int arow  = full ? (row0 + mr) : min(row0 + mr, nRows - 1);
    const float* xp = x + (size_t)arow * 128;

    v8f c = {};
    #pragma unroll 8
    for (int k0 = 0; k0 < 128; k0 += 4) {
        v2f a, b;
        a.x = xp[k0 + khalf];
        a.y = xp[k0 + khalf + 1];
        b.x = W1[(k0 + khalf) * 16 + mr];
        b.y = W1[(k0 + khalf + 1) * 16 + mr];
        c = wmma_f32(a, b, c);
    }
    int mbase = row0 + ((lane >> 4) << 3);
    float* yp = y + (size_t)mbase * 16 + mr;
    if (full) {
        #pragma unroll
        for (int v = 0; v < 8; ++v) yp[v * 16] = c[v];
    } else {
        #pragma unroll
        for (int v = 0; v < 8; ++v)
            if (mbase + v < nRows) yp[v * 16] = c[v];
    }
}

// -------- 16-wide scatter-add: acc[dst] += feat[src]  (+optional degree) ----
__global__ void k_scatter16(const float* __restrict__ feat, const int* __restrict__ src,
                            const int* __restrict__ dst, float* __restrict__ acc,
                            float* __restrict__ deg, int E) {
    int t = blockIdx.x * blockDim.x + threadIdx.x;
    int e = t >> 2;
    if (e >= E) return;
    int q = (t & 3) << 2;
    int d = dst[e];
    const float4 v = *(const float4*)(feat + (size_t)src[e] * 16 + q);
    float* ap = acc + (size_t)d * 16 + q;
    atomicAdd(ap + 0, v.x);
    atomicAdd(ap + 1, v.y);
    atomicAdd(ap + 2, v.z);
    atomicAdd(ap + 3, v.w);
    if (deg != nullptr && (t & 3) == 0) atomicAdd(&deg[d], 1.0f);
}

// ---------------- (neigh + self)/(deg+1) [+bias][relu] ----------------------
__global__ void k_finalize(const float* __restrict__ neigh, const float* __restrict__ selfv,
                           const float* __restrict__ deg, const float* __restrict__ bias,
                           float* __restrict__ out, int N, int hasBias, int hasRelu) {
    int t = blockIdx.x * blockDim.x + threadIdx.x;
    if (t >= N * 16) return;
    int i = t >> 4, f = t & 15;
    float s = (neigh[t] + selfv[t]) / (deg[i] + 1.0f);
    if (hasBias) s += bias[f];
    if (hasRelu) s = fmaxf(s, 0.0f);
    out[t] = s;
}

// ---------------- GEMM 2: h2[N,128] = agg2[N,16] @ W2[16,128] + b2 ----------
__global__ __launch_bounds__(256)
void k_gemm_aggW2(const float* __restrict__ A, const float* __restrict__ W2,
                  const float* __restrict__ b2, float* __restrict__ h2, int nRows) {
    int wave = (blockIdx.x * blockDim.x + threadIdx.x) >> 5;
    int lane = threadIdx.x & 31;
    int nTiles = (nRows + 15) >> 4;
    int rowTile = wave >> 3;           // 8 column tiles of 16 across the 128 outputs
    int colTile = wave & 7;
    if (rowTile >= nTiles) return;
    int row0 = rowTile << 4, col0 = colTile << 4;
    bool full = (row0 + 16 <= nRows);
    int mr = lane & 15;
    int khalf = (lane >> 4) << 1;
    int arow = full ? (row0 + mr) : min(row0 + mr, nRows - 1);
    const float* ap = A + (size_t)arow * 16;
    const float* Bp = W2 + col0;

    v8f c = {};
    #pragma unroll
    for (int k0 = 0; k0 < 16; k0 += 4) {
        v2f a, b;
        a.x = ap[k0 + khalf];
        a.y = ap[k0 + khalf + 1];
        b.x = Bp[(k0 + khalf) * 128 + mr];
        b.y = Bp[(k0 + khalf + 1) * 128 + mr];
        c = wmma_f32(a, b, c);
    }
    float bias = b2[col0 + mr];
    int mbase = row0 + ((lane >> 4) << 3);
    float* hp = h2 + (size_t)mbase * 128 + col0 + mr;
    if (full) {
        #pragma unroll
        for (int v = 0; v < 8; ++v) hp[v * 128] = c[v] + bias;
    } else {
        #pragma unroll
        for (int v = 0; v < 8; ++v)
            if (mbase + v < nRows) hp[v * 128] = c[v] + bias;
    }
}

// -------- GEMM 3: su = h2 @ Wmlp[0:128], sv = h2 @ Wmlp[128:256] ------------
__global__ __launch_bounds__(256)
void k_gemm_mlp(const float* __restrict__ h2, const float* __restrict__ Wmlp,
                float* __restrict__ su, float* __restrict__ sv, int nRows) {
    int wave = (blockIdx.x * blockDim.x + threadIdx.x) >> 5;
    int lane = threadIdx.x & 31;
    int nTiles = (nRows + 15) >> 4;
    if (wave >= nTiles) return;
    int row0 = wave << 4;
    bool full = (row0 + 16 <= nRows);
    int mr = lane & 15;
    int khalf = (lane >> 4) << 1;
    int arow = full ? (row0 + mr) : min(row0 + mr, nRows - 1);
    const float* ap = h2 + (size_t)arow * 128;
    const float* WA = Wmlp;              // rows 0..127   (stride 16)
    const float* WB = Wmlp + 128 * 16;   // rows 128..255

    v8f cu = {}, cv = {};
    #pragma unroll 4
    for (int k0 = 0; k0 < 128; k0 += 4) {
        v2f a, bu, bv;
        a.x  = ap[k0 + khalf];
        a.y  = ap[k0 + khalf + 1];
        bu.x = WA[(k0 + khalf) * 16 + mr];
        bu.y = WA[(k0 + khalf + 1) * 16 + mr];
        bv.x = WB[(k0 + khalf) * 16 + mr];
        bv.y = WB[(k0 + khalf + 1) * 16 + mr];
        cu = wmma_f32(a, bu, cu);        // shared A operand, two accumulators
        cv = wmma_f32(a, bv, cv);
    }
    int mbase = row0 + ((lane >> 4) << 3);
    float* up = su + (size_t)mbase * 16 + mr;
    float* vp = sv + (size_t)mbase * 16 + mr;
    if (full) {
        #pragma unroll
        for (int v = 0; v < 8; ++v) { up[v * 16] = cu[v]; vp[v * 16] = cv[v]; }
    } else {
        #pragma unroll
        for (int v = 0; v < 8; ++v)
            if (mbase + v < nRows) { up[v * 16] = cu[v]; vp[v * 16] = cv[v]; }
    }
}

// -------- per-edge: out[e] = su[src[e]] + sv[dst[e]] + bmlp -----------------
__global__ void k_edge(const float* __restrict__ su, const float* __restrict__ sv,
                       const int* __restrict__ src, const int* __restrict__ dst,
                       const float* __restrict__ bmlp, float* __restrict__ out, int E) {
    int t = blockIdx.x * blockDim.x + threadIdx.x;
    int e = t >> 2;
    if (e >= E) return;
    int q = (t & 3) << 2;
    const float4 u = *(const float4*)(su + (size_t)src[e] * 16 + q);
    const float4 v = *(const float4*)(sv + (size_t)dst[e] * 16 + q);
    const float4 b = *(const float4*)(bmlp + q);
    float4 r = make_float4(u.x + v.x + b.x, u.y + v.y + b.y,
                           u.z + v.z + b.z, u.w + v.w + b.w);
    *(float4*)(out + (size_t)e * 16 + q) = r;
}

extern "C" void kernel_launch(void* const* d_in, const int* in_sizes, int n_in,
                              void* d_out, int out_size, void* d_ws, size_t ws_size,
                              hipStream_t stream) {
    const float* x    = (const float*)d_in[0];
    const int*   src  = (const int*)d_in[1];
    const int*   dst  = (const int*)d_in[2];
    const float* W1   = (const float*)d_in[3];
    const float* b1   = (const float*)d_in[4];
    const float* W2   = (const float*)d_in[5];
    const float* b2   = (const float*)d_in[6];
    const float* Wmlp = (const float*)d_in[7];
    const float* bmlp = (const float*)d_in[8];
    float*       out  = (float*)d_out;

    const int N = in_sizes[0] / 128;   // 50000
    const int E = in_sizes[1];         // 800000

    float* ws = (float*)d_ws;
    const size_t n16 = (size_t)N * 16;
    float* y      = ws; ws += n16;
    float* neigh  = ws; ws += n16;
    float* h1     = ws; ws += n16;
    float* neigh2 = ws; ws += n16;
    float* agg2   = ws; ws += n16;
    float* su     = ws; ws += n16;
    float* sv     = ws; ws += n16;
    float* h2     = ws; ws += (size_t)N * 128;
    float* deg    = ws; ws += N;

    // zero the accumulation buffers every call (graph-capture safe)
    hipMemsetAsync(neigh,  0, n16 * sizeof(float), stream);
    hipMemsetAsync(neigh2, 0, n16 * sizeof(float), stream);
    hipMemsetAsync(deg,    0, (size_t)N * sizeof(float), stream);

    const int nTiles = (N + 15) / 16;                 // 3125
    const int T = 256;

    // layer 1 weight-first: y = x @ W1
    k_gemm_xW1<<<(nTiles + 7) / 8, T, 0, stream>>>(x, W1, y, N);
    // 16-dim aggregation of y, degree fused into this pass
    k_scatter16<<<(E * 4 + T - 1) / T, T, 0, stream>>>(y, src, dst, neigh, deg, E);
    // h1 = relu((neigh + y)/(deg+1) + b1)
    k_finalize<<<(N * 16 + T - 1) / T, T, 0, stream>>>(neigh, y, deg, b1, h1, N, 1, 1);
    // layer 2 aggregation of h1 (same graph -> same deg)
    k_scatter16<<<(E * 4 + T - 1) / T, T, 0, stream>>>(h1, src, dst, neigh2, nullptr, E);
    k_finalize<<<(N * 16 + T - 1) / T, T, 0, stream>>>(neigh2, h1, deg, b1, agg2, N, 0, 0);
    // h2 = agg2 @ W2 + b2
    k_gemm_aggW2<<<(nTiles * 8 + 7) / 8, T, 0, stream>>>(agg2, W2, b2, h2, N);
    // per-node predictor halves: su = h2 @ Wmlp_top, sv = h2 @ Wmlp_bot
    k_gemm_mlp<<<(nTiles + 7) / 8, T, 0, stream>>>(h2, Wmlp, su, sv, N);
    // per-edge gather-add (replaces 256x16 per-edge matmul)
    k_edge<<<(E * 4 + T - 1) / T, T, 0, stream>>>(su, sv, src, dst, bmlp, out, E);
}